// RootLoss_31198642438390
// MI455X (gfx1250) — compile-verified
//
#include <hip/hip_runtime.h>
#include <hip/hip_bf16.h>
#include <math.h>

// Problem constants (match reference)
#define Bc 2
#define Qc 2048
#define Fc 1538
#define FPAD 1552          // pad F to multiple of 16; zero triangles give solid angle exactly 0
#define NTILE (FPAD / 16)  // 97 f-tiles
#define ARRSTRIDE (Bc * FPAD)
// SoA scratch layout in d_ws (floats):
//  arrays 0..2: v0.xyz | 3..5: v1.xyz | 6..8: v2.xyz | 9..11: N = v1xv2+v2xv0+v0xv1
//  12: v0.v1 | 13: v1.v2 | 14: v0.v2 | 15: |v0|^2 | 16: |v1|^2 | 17: |v2|^2 | 18: det(v0,v1,v2)
#define RES_OFF   (19 * ARRSTRIDE)        // residuals (B*Q)
#define BATCH_OFF (RES_OFF + Bc * Qc)     // per-batch loss partials (B)
// total ws need: (BATCH_OFF + Bc) * 4 bytes ~= 252 KB

typedef __attribute__((ext_vector_type(2))) float v2f;
typedef __attribute__((ext_vector_type(8))) float v8f;

// ---------------------------------------------------------------------------
// Kernel 1: per-triangle invariants (and zero padding for f in [Fc, FPAD))
// ---------------------------------------------------------------------------
__global__ void wn_precompute_kernel(const float* __restrict__ tri, float* __restrict__ ws) {
  const int idx = blockIdx.x * blockDim.x + threadIdx.x;
  if (idx >= Bc * FPAD) return;
  const int b = idx / FPAD;
  const int f = idx - b * FPAD;

  float v[9];
  if (f < Fc) {
    const float* t = tri + ((size_t)(b * Fc + f)) * 9;
#pragma unroll
    for (int i = 0; i < 9; ++i) v[i] = t[i];
  } else {
#pragma unroll
    for (int i = 0; i < 9; ++i) v[i] = 0.0f;
  }
  const float v0x = v[0], v0y = v[1], v0z = v[2];
  const float v1x = v[3], v1y = v[4], v1z = v[5];
  const float v2x = v[6], v2y = v[7], v2z = v[8];

  const float c12x = v1y * v2z - v1z * v2y;
  const float c12y = v1z * v2x - v1x * v2z;
  const float c12z = v1x * v2y - v1y * v2x;
  const float c20x = v2y * v0z - v2z * v0y;
  const float c20y = v2z * v0x - v2x * v0z;
  const float c20z = v2x * v0y - v2y * v0x;
  const float c01x = v0y * v1z - v0z * v1y;
  const float c01y = v0z * v1x - v0x * v1z;
  const float c01z = v0x * v1y - v0y * v1x;

  float* S = ws + b * FPAD;
  S[ 0 * ARRSTRIDE + f] = v0x;
  S[ 1 * ARRSTRIDE + f] = v0y;
  S[ 2 * ARRSTRIDE + f] = v0z;
  S[ 3 * ARRSTRIDE + f] = v1x;
  S[ 4 * ARRSTRIDE + f] = v1y;
  S[ 5 * ARRSTRIDE + f] = v1z;
  S[ 6 * ARRSTRIDE + f] = v2x;
  S[ 7 * ARRSTRIDE + f] = v2y;
  S[ 8 * ARRSTRIDE + f] = v2z;
  S[ 9 * ARRSTRIDE + f] = c12x + c20x + c01x;
  S[10 * ARRSTRIDE + f] = c12y + c20y + c01y;
  S[11 * ARRSTRIDE + f] = c12z + c20z + c01z;
  S[12 * ARRSTRIDE + f] = v0x * v1x + v0y * v1y + v0z * v1z;
  S[13 * ARRSTRIDE + f] = v1x * v2x + v1y * v2y + v1z * v2z;
  S[14 * ARRSTRIDE + f] = v0x * v2x + v0y * v2y + v0z * v2z;
  S[15 * ARRSTRIDE + f] = v0x * v0x + v0y * v0y + v0z * v0z;
  S[16 * ARRSTRIDE + f] = v1x * v1x + v1y * v1y + v1z * v1z;
  S[17 * ARRSTRIDE + f] = v2x * v2x + v2y * v2y + v2z * v2z;
  S[18 * ARRSTRIDE + f] = v0x * c12x + v0y * c12y + v0z * c12z;
}

// ---------------------------------------------------------------------------
// Kernel 2: pairwise winding numbers via WMMA f32 16x16x4 + atan2 epilogue.
// Block = 128 threads (4 waves) = one 16-query tile; waves split f-tiles.
// A (16x4): lane-half h, VGPR v -> K = v + 2h (ISA 7.12.2); K=3 padded 0.
// B (4x16): same K mapping, N = lane&15. D: VGPR r -> row M = r + 8h, col N.
// ---------------------------------------------------------------------------
__global__ void __launch_bounds__(128) wn_pair_kernel(const float* __restrict__ pts,
                                                      const float* __restrict__ occ,
                                                      float* ws) {
  const int tid  = threadIdx.x;
  const int wave = tid >> 5;
  const int lane = tid & 31;
  const int half = lane >> 4;
  const int lcol = lane & 15;
  const int b    = blockIdx.x >> 7;          // 128 q-tiles per batch
  const int q0   = (blockIdx.x & 127) << 4;  // first query of tile

  __shared__ float pp[16];       // |p|^2 for the 16 queries of this tile
  __shared__ float wsum[4][16];  // per-wave row sums

  if (tid < 16) {
    const float* p = pts + ((size_t)(b * Qc + q0 + tid)) * 3;
    pp[tid] = p[0] * p[0] + p[1] * p[1] + p[2] * p[2];
  }

  // A matrix: rows = 16 query points, K = (x,y,z,0)
  const float* pa = pts + ((size_t)(b * Qc + q0 + lcol)) * 3;
  v2f A;
  if (half == 0) { A.x = pa[0]; A.y = pa[1]; }
  else           { A.x = pa[2]; A.y = 0.0f;  }

  __syncthreads();

  const float* S = ws + b * FPAD;  // SoA base for this batch

  float acc[8];
#pragma unroll
  for (int r = 0; r < 8; ++r) acc[r] = 0.0f;

  for (int ft = wave; ft < NTILE; ft += 4) {
    const int f0 = ft << 4;
    const int fb = f0 + lcol;

    if (f0 + 64 < FPAD) {
      // prefetch next f-tile's epilogue scalars (-> global_prefetch_b8)
      __builtin_prefetch(&S[12 * ARRSTRIDE + fb + 64], 0, 1);
      __builtin_prefetch(&S[0 * ARRSTRIDE + fb + 64], 0, 1);
    }

    // B matrices for v0, v1, v2, N (K = v + 2*half; K==3 -> zero pad row)
    v2f B0, B1, B2, BN;
    if (half == 0) {
      B0.x = S[ 0 * ARRSTRIDE + fb]; B0.y = S[ 1 * ARRSTRIDE + fb];
      B1.x = S[ 3 * ARRSTRIDE + fb]; B1.y = S[ 4 * ARRSTRIDE + fb];
      B2.x = S[ 6 * ARRSTRIDE + fb]; B2.y = S[ 7 * ARRSTRIDE + fb];
      BN.x = S[ 9 * ARRSTRIDE + fb]; BN.y = S[10 * ARRSTRIDE + fb];
    } else {
      B0.x = S[ 2 * ARRSTRIDE + fb]; B0.y = 0.0f;
      B1.x = S[ 5 * ARRSTRIDE + fb]; B1.y = 0.0f;
      B2.x = S[ 8 * ARRSTRIDE + fb]; B2.y = 0.0f;
      BN.x = S[11 * ARRSTRIDE + fb]; BN.y = 0.0f;
    }

    v8f z = {};
    v8f G0 = __builtin_amdgcn_wmma_f32_16x16x4_f32(false, A, false, B0, (short)0, z, false, false);
    v8f G1 = __builtin_amdgcn_wmma_f32_16x16x4_f32(false, A, false, B1, (short)0, z, false, false);
    v8f G2 = __builtin_amdgcn_wmma_f32_16x16x4_f32(false, A, false, B2, (short)0, z, false, false);
    v8f GN = __builtin_amdgcn_wmma_f32_16x16x4_f32(false, A, false, BN, (short)0, z, false, false);

    // Per-column (triangle) epilogue scalars: every one of this lane's 8
    // accumulator elements shares the same column N = fb.
    const float d01 = S[12 * ARRSTRIDE + fb];
    const float d12 = S[13 * ARRSTRIDE + fb];
    const float d02 = S[14 * ARRSTRIDE + fb];
    const float qa0 = S[15 * ARRSTRIDE + fb];
    const float qa1 = S[16 * ARRSTRIDE + fb];
    const float qa2 = S[17 * ARRSTRIDE + fb];
    const float det = S[18 * ARRSTRIDE + fb];

#pragma unroll
    for (int r = 0; r < 8; ++r) {
      const float ppr = pp[half * 8 + r];
      const float g0 = G0[r], g1 = G1[r], g2 = G2[r], gn = GN[r];
      const float n0 = sqrtf(fmaxf(qa0 - 2.0f * g0 + ppr, 0.0f));
      const float n1 = sqrtf(fmaxf(qa1 - 2.0f * g1 + ppr, 0.0f));
      const float n2 = sqrtf(fmaxf(qa2 - 2.0f * g2 + ppr, 0.0f));
      const float t01 = d01 - g0 - g1 + ppr;
      const float t12 = d12 - g1 - g2 + ppr;
      const float t02 = d02 - g0 - g2 + ppr;
      const float num = det - gn;
      const float den = n0 * n1 * n2 + t01 * n2 + t02 * n1 + t12 * n0;
      acc[r] += atan2f(num, den);  // solid angle / 2
    }
  }

  // Reduce each row across the 16 lanes of its half (wave32 shuffles).
#pragma unroll
  for (int r = 0; r < 8; ++r) {
    float v = acc[r];
    v += __shfl_xor(v, 1, 32);
    v += __shfl_xor(v, 2, 32);
    v += __shfl_xor(v, 4, 32);
    v += __shfl_xor(v, 8, 32);
    acc[r] = v;
  }
  if (lcol == 0) {
#pragma unroll
    for (int r = 0; r < 8; ++r) wsum[wave][half * 8 + r] = acc[r];
  }
  __syncthreads();
  if (tid < 16) {
    const float s = wsum[0][tid] + wsum[1][tid] + wsum[2][tid] + wsum[3][tid];
    const float wn = s * 0.15915494309189535f;  // (2*sum_atan2) / (4*pi)
    const int q = q0 + tid;
    ws[RES_OFF + b * Qc + q] = wn - occ[b * Qc + q];
  }
}

// ---------------------------------------------------------------------------
// Kernel 3: per-batch robust loss. LDS bitonic sort (lower median = s[1023]).
// ---------------------------------------------------------------------------
__device__ __forceinline__ void bitonic_sort_2048(float* s, int t) {
  for (unsigned k = 2; k <= 2048u; k <<= 1) {
    for (unsigned j = k >> 1; j > 0; j >>= 1) {
      __syncthreads();
      for (unsigned i = (unsigned)t; i < 2048u; i += 1024u) {
        const unsigned ixj = i ^ j;
        if (ixj > i) {
          const float a = s[i];
          const float c = s[ixj];
          const bool up = ((i & k) == 0u);
          const bool sw = up ? (a > c) : (a < c);
          if (sw) { s[i] = c; s[ixj] = a; }
        }
      }
    }
  }
  __syncthreads();
}

__global__ void __launch_bounds__(1024) robust_kernel(float* ws) {
  __shared__ float orig[2048];
  __shared__ float s[2048];
  const int b = blockIdx.x;
  const int t = threadIdx.x;
  const float* res = ws + RES_OFF + b * Qc;

  for (int i = t; i < 2048; i += 1024) {
    const float r = res[i];
    orig[i] = r;
    s[i] = r;
  }
  bitonic_sort_2048(s, t);          // leading barrier covers the fills
  const float med = s[1023];        // lower median, index (n-1)/2
  __syncthreads();
  for (int i = t; i < 2048; i += 1024) s[i] = fabsf(orig[i] - med);
  bitonic_sort_2048(s, t);
  const float mad = s[1023];
  __syncthreads();

  const float c = (mad / 0.67449f) * 4.6851f;  // robust_std * tune_const
  float local = 0.0f;
  for (int i = t; i < 2048; i += 1024) {
    const float r = orig[i];
    const float nr = r / c;
    const float om = 1.0f - nr * nr;
    const float w = (nr >= 1.0f) ? 0.0f : om * om;
    local += w * r * r;
  }
  s[t] = local;
  __syncthreads();
  for (int off = 512; off > 0; off >>= 1) {
    if (t < off) s[t] += s[t + off];
    __syncthreads();
  }
  if (t == 0) ws[BATCH_OFF + b] = s[0];
}

// ---------------------------------------------------------------------------
// Kernel 4: final scalar
// ---------------------------------------------------------------------------
__global__ void finalize_kernel(const float* __restrict__ ws, float* __restrict__ out) {
  if (threadIdx.x == 0 && blockIdx.x == 0) {
    out[0] = 0.5f * (ws[BATCH_OFF + 0] + ws[BATCH_OFF + 1]);
  }
}

extern "C" void kernel_launch(void* const* d_in, const int* in_sizes, int n_in,
                              void* d_out, int out_size, void* d_ws, size_t ws_size,
                              hipStream_t stream) {
  const float* pts = (const float*)d_in[0];  // (B,Q,3)
  const float* tri = (const float*)d_in[1];  // (B,F,3,3)
  const float* occ = (const float*)d_in[2];  // (B,Q)
  float* ws  = (float*)d_ws;                 // needs ~252 KB
  float* out = (float*)d_out;                // scalar loss (float32)
  (void)in_sizes; (void)n_in; (void)out_size; (void)ws_size;

  wn_precompute_kernel<<<(Bc * FPAD + 255) / 256, 256, 0, stream>>>(tri, ws);
  wn_pair_kernel<<<Bc * (Qc / 16), 128, 0, stream>>>(pts, occ, ws);
  robust_kernel<<<Bc, 1024, 0, stream>>>(ws);
  finalize_kernel<<<1, 1, 0, stream>>>(ws, out);
}